// DecoderRNN_86328842650311
// MI455X (gfx1250) — compile-verified
//
#include <hip/hip_runtime.h>
#include <math.h>

// Problem dims (fixed by the reference)
#define B_   64
#define T_   31
#define V_   32000
#define H_   512
#define E_   1024
#define L_   3
#define DD_  2
#define G4H_ 2048   // 4*H

typedef float v2f __attribute__((ext_vector_type(2)));
typedef float v8f __attribute__((ext_vector_type(8)));

__device__ __forceinline__ v8f v8zero() {
  v8f z;
#pragma unroll
  for (int i = 0; i < 8; ++i) z[i] = 0.0f;
  return z;
}

// Native fp32 WMMA: D(16x16) = A(16x4) * B(4x16) + C
__device__ __forceinline__ v8f wmma4(v2f a, v2f b, v8f c) {
  return __builtin_amdgcn_wmma_f32_16x16x4_f32(false, a, false, b, (short)0, c,
                                               false, false);
}

__device__ __forceinline__ float sigm(float x) {
  return 1.0f / (1.0f + __expf(-x));
}

// gfx1250 async copy: global -> LDS, 16 bytes per lane, tracked by ASYNCcnt.
__device__ __forceinline__ void async_cp16(unsigned int lds_off,
                                           const float* gaddr) {
  asm volatile("global_load_async_to_lds_b128 %0, %1, off"
               :
               : "v"(lds_off), "v"(gaddr)
               : "memory");
}

__device__ __forceinline__ unsigned int lds_off(const void* p) {
  // addrspace(3) offsets live in the low 32 bits of the generic pointer
  return (unsigned int)(unsigned long long)p;
}

// Shared tiling parameters for both GEMM kernels
#define PKC   32                 // K-chunk
#define PSTR  36                 // LDS row stride (floats): 16B-aligned, bank-conflict-free
#define A_FL  (B_ * PSTR)        // 2304 floats per A buffer
#define BW_FL (128 * PSTR)       // 4608 floats per B buffer

// ---------------------------------------------------------------- bias combine
__global__ void bias_combine_k(const float* __restrict__ bi,
                               const float* __restrict__ bh,
                               float* __restrict__ bc) {
  int i = blockIdx.x * blockDim.x + threadIdx.x;
  if (i < L_ * DD_ * G4H_) bc[i] = bi[i] + bh[i];
}

// ------------------------------------------------------- embed + relu gather
// xseq layout: [T][B][E]
__global__ __launch_bounds__(256) void embed_relu_k(const int* __restrict__ tgt,
                                                    const float* __restrict__ emb,
                                                    float* __restrict__ xseq) {
  int bt = blockIdx.x;
  int b = bt / T_;
  int t = bt % T_;
  int id = (t == 0) ? 0 : tgt[b * T_ + (t - 1)];  // SOS=0 else target[b, t-1]
  const float4* src = (const float4*)(emb + (size_t)id * E_);
  float4* dst = (float4*)(xseq + ((size_t)t * B_ + b) * E_);
  float4 v = src[threadIdx.x];  // 256 threads * 4 = 1024 = E
  v.x = fmaxf(v.x, 0.0f);
  v.y = fmaxf(v.y, 0.0f);
  v.z = fmaxf(v.z, 0.0f);
  v.w = fmaxf(v.w, 0.0f);
  dst[threadIdx.x] = v;
}

// ------------------------------------------------------------ fused LSTM layer
// gates = X @ Wih.T + hin @ Whh.T + bias, with X = [X0 | X1] (two [B,512]
// halves). K = 1536 processed as 48 chunks of 32 through double-buffered LDS
// filled with global_load_async_to_lds_b128. Block tile: 64 batch rows x
// (32 hidden cols x 4 gates) = 128 weight rows. After the GEMM each wave
// applies the LSTM nonlinearity and writes hout/cout for its 16x16 tile.
#define LCHK 48                  // 1536 / PKC

__global__ __launch_bounds__(256) void lstm_layer_k(
    const float* __restrict__ X0, const float* __restrict__ X1, int xstride,
    const float* __restrict__ hin_all, const float* __restrict__ cin_all,
    const float* __restrict__ Wih_all, const float* __restrict__ Whh_all,
    const float* __restrict__ bc_all, float* __restrict__ hout_all,
    float* __restrict__ cout_all, int layer) {
  __shared__ float Ab[2][A_FL];
  __shared__ float Bb[2][BW_FL];

  const int d = blockIdx.y;
  const int ld = layer * DD_ + d;
  const float* Wih = Wih_all + (size_t)ld * G4H_ * E_;   // [4H][E]
  const float* Whh = Whh_all + (size_t)ld * G4H_ * H_;   // [4H][H]
  const float* bias = bc_all + (size_t)ld * G4H_;        // [4H]
  const float* hin = hin_all + (size_t)ld * B_ * H_;
  const float* cin = cin_all + (size_t)ld * B_ * H_;
  float* hout = hout_all + (size_t)ld * B_ * H_;
  float* cout = cout_all + (size_t)ld * B_ * H_;

  const int tid = threadIdx.x;
  const int wave = tid >> 5;           // 0..7
  const int lane = tid & 31;
  const int rb = (wave & 3) * 16;      // batch-row tile base
  const int cg = (wave >> 2) * 16;     // col tile within the 32-wide slice
  const int l16 = lane & 15;
  const int hi = lane >> 4;            // K-pair select
  const int cbB = blockIdx.x * 32;     // block's hidden-col base (in [0,H))
  const int col = cbB + cg + l16;      // this lane's hidden column

  const unsigned int aoff = lds_off(&Ab[0][0]);
  const unsigned int boff = lds_off(&Bb[0][0]);

  v8f acc[4];
#pragma unroll
  for (int q = 0; q < 4; ++q) acc[q] = v8zero();

  // stage chunk c into buffer buf: A = 64 x 32 activations, B = 128 x 32
  // weight rows (4 gates x 32 cols of this block).
  auto stage = [&](int buf, int c) {
    const int kb = c * PKC;
    const float* Xs;
    int xs, xkb, wkb;
    const float* Wb;
    size_t wstr;
    if (c < 16) {
      Xs = X0; xs = xstride; Wb = Wih; wstr = E_; wkb = kb; xkb = kb;
    } else if (c < 32) {
      Xs = X1; xs = xstride; Wb = Wih; wstr = E_; wkb = kb; xkb = kb - 512;
    } else {
      Xs = hin; xs = H_; Wb = Whh; wstr = H_; wkb = kb - 1024; xkb = kb - 1024;
    }
#pragma unroll
    for (int i = 0; i < 2; ++i) {      // 512 float4 pieces of A
      int idx = tid + i * 256;
      int r = idx >> 3;
      int kl = (idx & 7) * 4;
      async_cp16(aoff + (unsigned)(buf * A_FL + r * PSTR + kl) * 4u,
                 Xs + (size_t)r * xs + xkb + kl);
    }
#pragma unroll
    for (int i = 0; i < 4; ++i) {      // 1024 float4 pieces of B
      int idx = tid + i * 256;
      int row = idx >> 3;              // 0..127 = gate*32 + colLocal
      int kl = (idx & 7) * 4;
      int q = row >> 5;
      int cl = row & 31;
      async_cp16(boff + (unsigned)(buf * BW_FL + row * PSTR + kl) * 4u,
                 Wb + (size_t)(q * H_ + cbB + cl) * wstr + wkb + kl);
    }
  };

  stage(0, 0);
  for (int c = 0; c < LCHK; ++c) {
    const int buf = c & 1;
    if (c + 1 < LCHK) {
      stage(buf ^ 1, c + 1);
      asm volatile("s_wait_asynccnt 0x6" ::: "memory");  // this stage done
    } else {
      asm volatile("s_wait_asynccnt 0x0" ::: "memory");
    }
    __syncthreads();

    const float* Al = &Ab[buf][0];
    const float* Bl = &Bb[buf][0];
#pragma unroll
    for (int kl = 0; kl < PKC; kl += 4) {
      const int kk = kl + 2 * hi;
      v2f a = *(const v2f*)(Al + (rb + l16) * PSTR + kk);
#pragma unroll
      for (int q = 0; q < 4; ++q) {
        v2f b = *(const v2f*)(Bl + (q * 32 + cg + l16) * PSTR + kk);
        acc[q] = wmma4(a, b, acc[q]);
      }
    }
    __syncthreads();  // buffer reuse fence
  }

  float bq[4];
#pragma unroll
  for (int q = 0; q < 4; ++q) bq[q] = bias[q * H_ + col];

  // D layout: acc[j] holds (M = rb + j + 8*hi, N = col)
#pragma unroll
  for (int j = 0; j < 8; ++j) {
    int brow = rb + j + 8 * hi;
    int idx = brow * H_ + col;
    float iv = sigm(acc[0][j] + bq[0]);
    float fv = sigm(acc[1][j] + bq[1]);
    float gv = tanhf(acc[2][j] + bq[2]);
    float ov = sigm(acc[3][j] + bq[3]);
    float cn = fv * cin[idx] + iv * gv;
    float hn = ov * tanhf(cn);
    cout[idx] = cn;
    hout[idx] = hn;
  }
}

// -------------------------------------------------------- vocab projection
// logits[b, v] = concat(P0,P1)[b,:] . out_W[v,:] + out_b[v]
// Block tile: 64 (all batch rows) x 128 vocab cols; K pipelined in chunks of 32
// through double-buffered LDS filled with global_load_async_to_lds_b128.
#define NCHK  (E_ / PKC)         // 32 chunks

__global__ __launch_bounds__(256) void proj_k(const float* __restrict__ P0,
                                              const float* __restrict__ P1,
                                              const float* __restrict__ Wo,
                                              const float* __restrict__ bo,
                                              float* __restrict__ out, int t) {
  __shared__ float Ab[2][A_FL];
  __shared__ float Bb[2][BW_FL];

  const int tid = threadIdx.x;
  const int wave = tid >> 5;
  const int lane = tid & 31;
  const int rb = (wave & 3) * 16;        // batch-row tile base
  const int ng = wave >> 2;              // col half: 0 or 1 (64 cols each)
  const int l16 = lane & 15;
  const int hi = lane >> 4;
  const int vbase = blockIdx.x * 128;    // block's vocab base

  const unsigned int aoff = lds_off(&Ab[0][0]);
  const unsigned int boff = lds_off(&Bb[0][0]);

  v8f acc[4];
#pragma unroll
  for (int n = 0; n < 4; ++n) acc[n] = v8zero();

  auto stage = [&](int buf, int c) {
    const int kb = c * PKC;
    const float* X = (kb < 512) ? (P0 + kb) : (P1 + (kb - 512));
#pragma unroll
    for (int i = 0; i < 2; ++i) {        // 512 float4 pieces of A
      int idx = tid + i * 256;
      int r = idx >> 3;
      int kl = (idx & 7) * 4;
      async_cp16(aoff + (unsigned)(buf * A_FL + r * PSTR + kl) * 4u,
                 X + (size_t)r * H_ + kl);
    }
    const float* W = Wo + kb;
#pragma unroll
    for (int i = 0; i < 4; ++i) {        // 1024 float4 pieces of B
      int idx = tid + i * 256;
      int vr = idx >> 3;
      int kl = (idx & 7) * 4;
      async_cp16(boff + (unsigned)(buf * BW_FL + vr * PSTR + kl) * 4u,
                 W + (size_t)(vbase + vr) * E_ + kl);
    }
  };

  stage(0, 0);
  for (int c = 0; c < NCHK; ++c) {
    const int buf = c & 1;
    if (c + 1 < NCHK) {
      stage(buf ^ 1, c + 1);
      // 6 newest async ops belong to the next stage; wait for this stage's.
      asm volatile("s_wait_asynccnt 0x6" ::: "memory");
    } else {
      asm volatile("s_wait_asynccnt 0x0" ::: "memory");
    }
    __syncthreads();

    const float* Al = &Ab[buf][0];
    const float* Bl = &Bb[buf][0];
#pragma unroll
    for (int kl = 0; kl < PKC; kl += 4) {
      const int kk = kl + 2 * hi;
      v2f a = *(const v2f*)(Al + (rb + l16) * PSTR + kk);
#pragma unroll
      for (int n = 0; n < 4; ++n) {
        v2f b = *(const v2f*)(Bl + (ng * 64 + n * 16 + l16) * PSTR + kk);
        acc[n] = wmma4(a, b, acc[n]);
      }
    }
    __syncthreads();  // buffer `buf` is re-filled in the next iteration
  }

#pragma unroll
  for (int n = 0; n < 4; ++n) {
    int vcol = vbase + ng * 64 + n * 16 + l16;
    float bv = bo[vcol];
#pragma unroll
    for (int j = 0; j < 8; ++j) {
      int brow = rb + j + 8 * hi;
      out[((size_t)brow * T_ + t) * V_ + vcol] = acc[n][j] + bv;
    }
  }
}

// ------------------------------------------------------ in-place log-softmax
__global__ __launch_bounds__(256) void logsoftmax_k(float* __restrict__ out) {
  __shared__ float red[256];
  float* row = out + (size_t)blockIdx.x * V_;  // rows are contiguous [(b*T+t)*V]

  float m = -INFINITY;
  for (int i = threadIdx.x; i < V_; i += 256) m = fmaxf(m, row[i]);
  red[threadIdx.x] = m;
  __syncthreads();
  for (int s = 128; s > 0; s >>= 1) {
    if (threadIdx.x < s) red[threadIdx.x] = fmaxf(red[threadIdx.x], red[threadIdx.x + s]);
    __syncthreads();
  }
  m = red[0];
  __syncthreads();

  float ssum = 0.0f;
  for (int i = threadIdx.x; i < V_; i += 256) ssum += __expf(row[i] - m);
  red[threadIdx.x] = ssum;
  __syncthreads();
  for (int s = 128; s > 0; s >>= 1) {
    if (threadIdx.x < s) red[threadIdx.x] += red[threadIdx.x + s];
    __syncthreads();
  }
  float lse = m + __logf(red[0]);
  __syncthreads();

  for (int i = threadIdx.x; i < V_; i += 256) row[i] -= lse;
}

// ---------------------------------------------------------------- launcher
extern "C" void kernel_launch(void* const* d_in, const int* in_sizes, int n_in,
                              void* d_out, int out_size, void* d_ws,
                              size_t ws_size, hipStream_t stream) {
  (void)in_sizes; (void)n_in; (void)out_size; (void)ws_size;
  const float* enc_h = (const float*)d_in[1];
  const float* enc_c = (const float*)d_in[2];
  const int* tgt = (const int*)d_in[3];
  const float* emb = (const float*)d_in[4];
  const float* Wih = (const float*)d_in[5];
  const float* Whh = (const float*)d_in[6];
  const float* bih = (const float*)d_in[7];
  const float* bhh = (const float*)d_in[8];
  const float* Wo = (const float*)d_in[9];
  const float* bo = (const float*)d_in[10];
  float* out = (float*)d_out;
  float* ws = (float*)d_ws;

  const size_t ST = (size_t)L_ * DD_ * B_ * H_;  // 196608 floats per state buf
  float* xseq = ws;                               // [T][B][E]
  float* h0 = xseq + (size_t)T_ * B_ * E_;
  float* c0 = h0 + ST;
  float* h1 = c0 + ST;
  float* c1 = h1 + ST;
  float* bc = c1 + ST;                            // combined bias [L][D][4H]

  hipMemcpyAsync(h0, enc_h, ST * sizeof(float), hipMemcpyDeviceToDevice, stream);
  hipMemcpyAsync(c0, enc_c, ST * sizeof(float), hipMemcpyDeviceToDevice, stream);

  bias_combine_k<<<(L_ * DD_ * G4H_ + 255) / 256, 256, 0, stream>>>(bih, bhh, bc);
  embed_relu_k<<<B_ * T_, 256, 0, stream>>>(tgt, emb, xseq);

  for (int t = 0; t < T_; ++t) {
    float* hin = (t & 1) ? h1 : h0;
    float* cin = (t & 1) ? c1 : c0;
    float* hout = (t & 1) ? h0 : h1;
    float* cout = (t & 1) ? c0 : c1;
    for (int l = 0; l < L_; ++l) {
      const float* X0;
      const float* X1;
      int xst;
      if (l == 0) {
        X0 = xseq + (size_t)t * B_ * E_;
        X1 = X0 + 512;
        xst = E_;
      } else {
        // concat of previous layer's fresh outputs (both directions)
        X0 = hout + (size_t)((l - 1) * DD_ + 0) * B_ * H_;
        X1 = hout + (size_t)((l - 1) * DD_ + 1) * B_ * H_;
        xst = H_;
      }
      lstm_layer_k<<<dim3(H_ / 32, DD_), 256, 0, stream>>>(
          X0, X1, xst, hin, cin, Wih, Whh, bc, hout, cout, l);
    }
    const float* P0 = hout + (size_t)(2 * DD_ + 0) * B_ * H_;
    const float* P1 = hout + (size_t)(2 * DD_ + 1) * B_ * H_;
    proj_k<<<V_ / 128, 256, 0, stream>>>(P0, P1, Wo, bo, out, t);
  }

  logsoftmax_k<<<B_ * T_, 256, 0, stream>>>(out);

  // t = 30 is even -> final states live in buffer set 1
  hipMemcpyAsync(out + (size_t)B_ * T_ * V_, h1, ST * sizeof(float),
                 hipMemcpyDeviceToDevice, stream);
  hipMemcpyAsync(out + (size_t)B_ * T_ * V_ + ST, c1, ST * sizeof(float),
                 hipMemcpyDeviceToDevice, stream);
}